// DotAttention_9947144257895
// MI455X (gfx1250) — compile-verified
//
#include <hip/hip_runtime.h>
#include <math.h>

// Problem constants (from reference): B=32, S=2048, H=1024, F=2H=2048
#define Bn 32
#define Sn 2048
#define Hn 1024
#define Fn 2048
#define SCHUNK 512            // S split into 4 chunks for the weighted-sum pass

typedef float v2f __attribute__((ext_vector_type(2)));
typedef float v8f __attribute__((ext_vector_type(8)));

// ---------------------------------------------------------------------------
// c[b] = b_enc . dh[b]
// ---------------------------------------------------------------------------
__global__ void k_cvec(const float* __restrict__ dh, const float* __restrict__ b_enc,
                       float* __restrict__ c) {
    int b = blockIdx.x, t = threadIdx.x;
    __shared__ float red[256];
    float acc = 0.f;
    for (int h = t; h < Hn; h += 256) acc += b_enc[h] * dh[b * Hn + h];
    red[t] = acc; __syncthreads();
    for (int o = 128; o; o >>= 1) { if (t < o) red[t] += red[t + o]; __syncthreads(); }
    if (!t) c[b] = red[0];
}

// ---------------------------------------------------------------------------
// U[b][f] = sum_h W_enc[f][h] * dh[b][h]   (M=f tile 16, N=b tile 16, K=4 WMMA f32)
// 128 f-tiles x 2 b-tiles = 256 wave-tiles, 4 waves/block
// ---------------------------------------------------------------------------
__global__ void k_proj_u(const float* __restrict__ W_enc, const float* __restrict__ dh,
                         float* __restrict__ U) {
    int lane = threadIdx.x & 31;
    int tile = blockIdx.x * 4 + (threadIdx.x >> 5);   // 0..255
    int f0 = (tile >> 1) * 16;
    int b0 = (tile & 1) * 16;
    int m  = lane & 15;          // A: row M ; B: col N ; D: col N
    int hi = lane >> 4;
    v8f acc = {};
    for (int k0 = 0; k0 < Hn; k0 += 4) {
        int ka = k0 + 2 * hi;    // lanes 0-15: K={0,1}; lanes 16-31: K={2,3}
        v2f a  = *(const v2f*)(W_enc + (size_t)(f0 + m) * Hn + ka);
        v2f bb = *(const v2f*)(dh    + (size_t)(b0 + m) * Hn + ka);
        acc = __builtin_amdgcn_wmma_f32_16x16x4_f32(false, a, false, bb,
                                                    (short)0, acc, false, false);
    }
    #pragma unroll
    for (int i = 0; i < 8; ++i) {
        int mm = i + 8 * hi;     // D: VGPR i -> M=i (lanes 0-15) / M=i+8 (lanes 16-31)
        U[(size_t)(b0 + m) * Fn + (f0 + mm)] = acc[i];
    }
}

// ---------------------------------------------------------------------------
// Streaming pass 1: raw scores(b,s) = enc(b,s,:) . U[b] + c[b]
// One wave per row; b128 coalesced loads; wave32 shuffle reduction.
// ---------------------------------------------------------------------------
__global__ void k_scores(const float* __restrict__ enc, const float* __restrict__ U,
                         const float* __restrict__ c, float* __restrict__ attn) {
    int wave = (int)((blockIdx.x * blockDim.x + threadIdx.x) >> 5);
    int lane = threadIdx.x & 31;
    if (wave >= Bn * Sn) return;
    int b = wave >> 11;                               // / S
    const float4* e4 = (const float4*)(enc + (size_t)wave * Fn);
    const float4* u4 = (const float4*)(U + (size_t)b * Fn);
    float acc = 0.f;
    #pragma unroll
    for (int i = 0; i < 16; ++i) {                    // 2048 floats / 32 lanes / 4
        float4 ev = e4[i * 32 + lane];
        float4 uv = u4[i * 32 + lane];
        acc += ev.x * uv.x + ev.y * uv.y + ev.z * uv.z + ev.w * uv.w;
    }
    for (int o = 16; o; o >>= 1) acc += __shfl_xor(acc, o, 32);
    if (!lane) attn[wave] = acc + c[b];
}

// ---------------------------------------------------------------------------
// In-place softmax over S per batch (one block per b, 8 elems/thread)
// ---------------------------------------------------------------------------
__global__ void k_softmax(float* __restrict__ attn) {
    int b = blockIdx.x, t = threadIdx.x;
    __shared__ float red[256];
    float v[8];
    float mx = -3.402823466e38f;
    #pragma unroll
    for (int i = 0; i < 8; ++i) { v[i] = attn[b * Sn + i * 256 + t]; mx = fmaxf(mx, v[i]); }
    red[t] = mx; __syncthreads();
    for (int o = 128; o; o >>= 1) { if (t < o) red[t] = fmaxf(red[t], red[t + o]); __syncthreads(); }
    mx = red[0]; __syncthreads();
    float s = 0.f;
    #pragma unroll
    for (int i = 0; i < 8; ++i) { v[i] = expf(v[i] - mx); s += v[i]; }
    red[t] = s; __syncthreads();
    for (int o = 128; o; o >>= 1) { if (t < o) red[t] += red[t + o]; __syncthreads(); }
    float inv = 1.f / red[0];
    #pragma unroll
    for (int i = 0; i < 8; ++i) attn[b * Sn + i * 256 + t] = v[i] * inv;
}

// ---------------------------------------------------------------------------
// Streaming pass 2 (partial): Epart[c][b][f] = sum_{s in chunk c} w(b,s)*enc(b,s,f)
// grid (Fn/1024, Bn, Sn/SCHUNK) = (2,32,4); block 256; each lane loads float4
// -> 4KB contiguous burst per step per block; weights cached in LDS;
// global_prefetch_b8 keeps the HBM pipe full.
// ---------------------------------------------------------------------------
__global__ void k_esum(const float* __restrict__ enc, const float* __restrict__ attn,
                       float* __restrict__ Epart) {
    int b  = blockIdx.y;
    int cc = blockIdx.z;
    int f  = blockIdx.x * 1024 + threadIdx.x * 4;
    __shared__ float wsm[SCHUNK];                      // 2 KB of LDS
    int s0 = cc * SCHUNK;
    for (int i = threadIdx.x; i < SCHUNK; i += 256) wsm[i] = attn[b * Sn + s0 + i];
    __syncthreads();
    const float4* ep = (const float4*)(enc + ((size_t)(b * Sn + s0)) * Fn + f);
    const size_t rs = Fn / 4;                          // row stride in float4s
    float4 acc = make_float4(0.f, 0.f, 0.f, 0.f);
    for (int s = 0; s < SCHUNK; ++s) {
        int sp = s + 8; if (sp > SCHUNK - 1) sp = SCHUNK - 1;
        __builtin_prefetch(ep + (size_t)sp * rs, 0, 0);
        float4 ev = ep[(size_t)s * rs];
        float w = wsm[s];
        acc.x += w * ev.x; acc.y += w * ev.y; acc.z += w * ev.z; acc.w += w * ev.w;
    }
    *(float4*)(Epart + ((size_t)cc * Bn + b) * Fn + f) = acc;
}

// ---------------------------------------------------------------------------
// E[b][f] = sum_c Epart[c][b][f]  (deterministic reduction, b128 loads/stores)
// ---------------------------------------------------------------------------
__global__ void k_ereduce(const float* __restrict__ Epart, float* __restrict__ E) {
    int idx = (blockIdx.x * 256 + threadIdx.x) * 4;    // over B*F = 65536 floats
    const size_t stride = (size_t)Bn * Fn;
    float4 a0 = *(const float4*)(Epart + idx);
    float4 a1 = *(const float4*)(Epart + stride + idx);
    float4 a2 = *(const float4*)(Epart + 2 * stride + idx);
    float4 a3 = *(const float4*)(Epart + 3 * stride + idx);
    float4 r = make_float4(a0.x + a1.x + a2.x + a3.x,
                           a0.y + a1.y + a2.y + a3.y,
                           a0.z + a1.z + a2.z + a3.z,
                           a0.w + a1.w + a2.w + a3.w);
    *(float4*)(E + idx) = r;
}

// ---------------------------------------------------------------------------
// CTX[b][j] = sum_f E[b][f] * W_enc[f][j] + b_enc[j]   (M=32, N=1024, K=2048)
// 2 m-tiles x 64 n-tiles = 128 wave-tiles, 4 waves/block, WMMA f32 16x16x4
// ---------------------------------------------------------------------------
__global__ void k_context(const float* __restrict__ E, const float* __restrict__ W_enc,
                          const float* __restrict__ b_enc, float* __restrict__ CTX) {
    int lane = threadIdx.x & 31;
    int tile = blockIdx.x * 4 + (threadIdx.x >> 5);    // 0..127
    int m0 = (tile & 1) * 16;
    int n0 = (tile >> 1) * 16;
    int m  = lane & 15;
    int hi = lane >> 4;
    v8f acc = {};
    for (int k0 = 0; k0 < Fn; k0 += 4) {
        int ka = k0 + 2 * hi;
        v2f a = *(const v2f*)(E + (size_t)(m0 + m) * Fn + ka);
        v2f bb;
        bb.x = W_enc[(size_t)ka * Hn + (n0 + m)];
        bb.y = W_enc[(size_t)(ka + 1) * Hn + (n0 + m)];
        acc = __builtin_amdgcn_wmma_f32_16x16x4_f32(false, a, false, bb,
                                                    (short)0, acc, false, false);
    }
    #pragma unroll
    for (int i = 0; i < 8; ++i) {
        int mm = i + 8 * hi;
        CTX[(size_t)(m0 + mm) * Hn + (n0 + m)] = acc[i] + b_enc[n0 + m];
    }
}

// ---------------------------------------------------------------------------
// out[b][h] = tanh( [CTX | dh](b,:) @ W_ctx + b_ctx )   (M=32, N=1024, K=2048)
// ---------------------------------------------------------------------------
__global__ void k_out(const float* __restrict__ CTX, const float* __restrict__ dh,
                      const float* __restrict__ W_ctx, const float* __restrict__ b_ctx,
                      float* __restrict__ out) {
    int lane = threadIdx.x & 31;
    int tile = blockIdx.x * 4 + (threadIdx.x >> 5);    // 0..127
    int m0 = (tile & 1) * 16;
    int n0 = (tile >> 1) * 16;
    int m  = lane & 15;
    int hi = lane >> 4;
    v8f acc = {};
    for (int k0 = 0; k0 < Fn; k0 += 4) {
        int ka = k0 + 2 * hi;
        v2f a;
        a.x = (ka     < Hn) ? CTX[(size_t)(m0 + m) * Hn + ka]
                            : dh [(size_t)(m0 + m) * Hn + (ka - Hn)];
        a.y = (ka + 1 < Hn) ? CTX[(size_t)(m0 + m) * Hn + ka + 1]
                            : dh [(size_t)(m0 + m) * Hn + (ka + 1 - Hn)];
        v2f bb;
        bb.x = W_ctx[(size_t)ka * Hn + (n0 + m)];
        bb.y = W_ctx[(size_t)(ka + 1) * Hn + (n0 + m)];
        acc = __builtin_amdgcn_wmma_f32_16x16x4_f32(false, a, false, bb,
                                                    (short)0, acc, false, false);
    }
    #pragma unroll
    for (int i = 0; i < 8; ++i) {
        int mm = i + 8 * hi;
        out[(size_t)(m0 + mm) * Hn + (n0 + m)] = tanhf(acc[i] + b_ctx[n0 + m]);
    }
}

// ---------------------------------------------------------------------------
extern "C" void kernel_launch(void* const* d_in, const int* in_sizes, int n_in,
                              void* d_out, int out_size, void* d_ws, size_t ws_size,
                              hipStream_t stream) {
    const float* enc   = (const float*)d_in[0];   // (B,S,2H)
    const float* dh    = (const float*)d_in[1];   // (1,B,H) -> (B,H)
    const float* W_enc = (const float*)d_in[2];   // (2H,H)
    const float* b_enc = (const float*)d_in[3];   // (H,)
    const float* W_ctx = (const float*)d_in[4];   // (2H,H)
    const float* b_ctx = (const float*)d_in[5];   // (H,)

    float* out  = (float*)d_out;                  // (B,1,H) = 32768 floats
    float* attn = out + (size_t)Bn * Hn;          // (B,S,1) = 65536 floats

    // ws layout (floats): U(B*F) | c(B) | CTX(B*H) | E(B*F) | Epart(4*B*F) (~1.7 MB)
    float* U     = (float*)d_ws;
    float* c     = U + (size_t)Bn * Fn;
    float* CTX   = c + Bn;
    float* E     = CTX + (size_t)Bn * Hn;
    float* Epart = E + (size_t)Bn * Fn;

    k_cvec   <<<Bn, 256, 0, stream>>>(dh, b_enc, c);
    k_proj_u <<<64, 128, 0, stream>>>(W_enc, dh, U);                       // 256 tiles
    k_scores <<<(Bn * Sn * 32) / 256, 256, 0, stream>>>(enc, U, c, attn);  // 65536 waves
    k_softmax<<<Bn, 256, 0, stream>>>(attn);
    k_esum   <<<dim3(Fn / 1024, Bn, Sn / SCHUNK), 256, 0, stream>>>(enc, attn, Epart);
    k_ereduce<<<(Bn * Fn) / (256 * 4), 256, 0, stream>>>(Epart, E);
    k_context<<<32, 128, 0, stream>>>(E, W_enc, b_enc, CTX);               // 128 tiles
    k_out    <<<32, 128, 0, stream>>>(CTX, dh, W_ctx, b_ctx, out);         // 128 tiles
}